// DGCNNSeal_84086869721209
// MI455X (gfx1250) — compile-verified
//
#include <hip/hip_runtime.h>
#include <math.h>

// ---------------------------------------------------------------------------
// DGCNN/SEAL forward for MI455X (gfx1250, wave32).
//   GCN dense GEMMs      -> V_WMMA_F32_16X16X4_F32, branch-free b64 fragment
//                           loads (padded A, transposed+padded B), uniform-
//                           guarded store epilogue
//   edge scatter/gather  -> wave-per-edge, lane-per-feature, global f32 atomics
//                           (hw/agg are 12.8 MB each -> L2-resident)
//   sort-pool            -> O(n_g^2) rank (exact lexsort tie semantics)
//   conv/mlp head        -> one workgroup per graph, LDS staged
// ---------------------------------------------------------------------------

typedef float v2f __attribute__((ext_vector_type(2)));
typedef float v8f __attribute__((ext_vector_type(8)));

#define TPB 256

// ----------------------------- utility ------------------------------------
__global__ void k_zero_f(float* __restrict__ p, long n) {
  long i = (long)blockIdx.x * blockDim.x + threadIdx.x;
  if (i < n) p[i] = 0.f;
}
__global__ void k_zero_i(int* __restrict__ p, int n) {
  int i = blockIdx.x * blockDim.x + threadIdx.x;
  if (i < n) p[i] = 0;
}

// ----------------------------- degree / norms ------------------------------
__global__ void k_deg(const int* __restrict__ ei, int* __restrict__ degi, int E) {
  int e = blockIdx.x * blockDim.x + threadIdx.x;
  if (e < E) atomicAdd(&degi[ei[E + e]], 1);
}
__global__ void k_dinv(const int* __restrict__ degi, float* __restrict__ dinv,
                       float* __restrict__ snorm, int n) {
  int i = blockIdx.x * blockDim.x + threadIdx.x;
  if (i >= n) return;
  float d = rsqrtf(1.0f + (float)degi[i]);
  dinv[i] = d;
  snorm[i] = d * d;
}
__global__ void k_enorm(const int* __restrict__ ei, const float* __restrict__ dinv,
                        float* __restrict__ enorm, int E) {
  int e = blockIdx.x * blockDim.x + threadIdx.x;
  if (e < E) enorm[e] = dinv[ei[e]] * dinv[ei[E + e]];
}

// ----------------------------- GEMM prep -----------------------------------
// Zero-pad x from 14 -> 16 columns.
__global__ void k_pad_x(const float* __restrict__ x, float* __restrict__ xp, int n) {
  int idx = blockIdx.x * blockDim.x + threadIdx.x;
  if (idx >= n * 16) return;
  int i = idx >> 4, f = idx & 15;
  xp[idx] = (f < 14) ? x[(long)i * 14 + f] : 0.f;
}
// Transpose W[Kreal x 32] (row-major) into Wt[32 x KP] column-major, K zero-padded.
__global__ void k_prep_w(const float* __restrict__ W, float* __restrict__ Wt,
                         int Kreal, int KP) {
  int idx = blockIdx.x * blockDim.x + threadIdx.x;
  if (idx >= 32 * KP) return;
  int c = idx / KP, k = idx % KP;
  Wt[idx] = (k < Kreal) ? W[k * 32 + c] : 0.f;
}

// ----------------------------- WMMA GEMM -----------------------------------
// out[N x 32] = A[N x KP] @ W, f32 WMMA 16x16x4. One wave per 16x16 tile.
// A lane layout (16x4 f32): lanes 0-15 hold K=k0,k0+1 for M=lane; lanes 16-31
// hold K=k0+2,k0+3 for M=lane-16. B mirrored (per-lane col, consecutive K).
// C/D: VGPR r, lane L -> C[r + (L>=16)*8][L%16].
// Branch-free loads (padded A, transposed+padded B, clamped tail rows);
// wave-uniform full-tile test keeps the store epilogue unconditional.
template <int KP>
__global__ void k_gemm32_wmma(const float* __restrict__ A,   // N x KP row-major
                              const float* __restrict__ Wt,  // 32 x KP col-major
                              float* __restrict__ out, int n) {
  int wave = (int)(((long)blockIdx.x * blockDim.x + threadIdx.x) >> 5);
  int lane = threadIdx.x & 31;
  int mtiles = (n + 15) >> 4;
  if (wave >= mtiles * 2) return;           // wave-uniform exit
  int mtile = wave >> 1;
  int ntile = wave & 1;
  int half  = lane >> 4;                    // 0 or 1
  int l16   = lane & 15;
  int row   = mtile * 16 + l16;
  int rowc  = row < n ? row : n - 1;        // clamp loads; stores guarded below
  int col   = ntile * 16 + l16;
  const float* arow = A + (long)rowc * KP + half * 2;
  const float* bcol = Wt + col * KP + half * 2;

  v8f c = {0.f, 0.f, 0.f, 0.f, 0.f, 0.f, 0.f, 0.f};
#pragma unroll
  for (int k0 = 0; k0 < KP; k0 += 4) {
    v2f a = *(const v2f*)(arow + k0);       // global_load_b64
    v2f b = *(const v2f*)(bcol + k0);       // global_load_b64
    c = __builtin_amdgcn_wmma_f32_16x16x4_f32(false, a, false, b,
                                              (short)0, c, false, false);
  }

  int orow0 = mtile * 16 + half * 8;        // this lane's first output row
  float* obase = out + (long)orow0 * 32 + col;
  if (mtile * 16 + 16 <= n) {               // wave-uniform: full tile (hot path)
#pragma unroll
    for (int r = 0; r < 8; ++r) obase[(long)r * 32] = c[r];
  } else {                                  // tail tile only
#pragma unroll
    for (int r = 0; r < 8; ++r)
      if (orow0 + r < n) obase[(long)r * 32] = c[r];
  }
}

// ----------------------------- edge aggregation ----------------------------
// One wave per edge; lane = feature. 128B coalesced gather + 128B atomic add.
__global__ void k_edge_agg32(const int* __restrict__ ei, const float* __restrict__ enorm,
                             const float* __restrict__ hw, float* __restrict__ agg, int E) {
  long t = (long)blockIdx.x * blockDim.x + threadIdx.x;
  int e = (int)(t >> 5);
  int lane = threadIdx.x & 31;
  if (e >= E) return;
  int s = ei[e], d = ei[E + e];
  float v = hw[(long)s * 32 + lane] * enorm[e];
  unsafeAtomicAdd(&agg[(long)d * 32 + lane], v);
}

// h_next = tanh(agg + hw*self_norm + b); also scatter into per_atom columns.
__global__ void k_combine(const float* __restrict__ agg, const float* __restrict__ hw,
                          const float* __restrict__ snorm, const float* __restrict__ b,
                          float* __restrict__ hcur, float* __restrict__ per_atom,
                          int coloff, int n) {
  int idx = blockIdx.x * blockDim.x + threadIdx.x;
  if (idx >= n * 32) return;
  int i = idx >> 5, f = idx & 31;
  float v = tanhf(agg[idx] + hw[idx] * snorm[i] + b[f]);
  hcur[idx] = v;
  per_atom[(long)i * 97 + coloff + f] = v;
}

// ----------------------------- layer 3 (H -> 1) ----------------------------
__global__ void k_hw4(const float* __restrict__ h, const float* __restrict__ W3,
                      float* __restrict__ hw4, int n) {
  int i = blockIdx.x * blockDim.x + threadIdx.x;
  if (i >= n) return;
  float acc = 0.f;
#pragma unroll
  for (int f = 0; f < 32; ++f) acc += h[(long)i * 32 + f] * W3[f];
  hw4[i] = acc;
}
__global__ void k_edge_agg1(const int* __restrict__ ei, const float* __restrict__ enorm,
                            const float* __restrict__ hw4, float* __restrict__ agg4, int E) {
  int e = blockIdx.x * blockDim.x + threadIdx.x;
  if (e >= E) return;
  unsafeAtomicAdd(&agg4[ei[E + e]], hw4[ei[e]] * enorm[e]);
}
__global__ void k_combine4(const float* __restrict__ agg4, const float* __restrict__ hw4,
                           const float* __restrict__ snorm, const float* __restrict__ b3,
                           float* __restrict__ per_atom, int n) {
  int i = blockIdx.x * blockDim.x + threadIdx.x;
  if (i >= n) return;
  per_atom[(long)i * 97 + 96] = tanhf(agg4[i] + hw4[i] * snorm[i] + b3[0]);
}

// ----------------------------- sort pooling --------------------------------
__global__ void k_counts(const int* __restrict__ batch, int* __restrict__ counts, int n) {
  int i = blockIdx.x * blockDim.x + threadIdx.x;
  if (i < n) atomicAdd(&counts[batch[i]], 1);
}
__global__ void k_starts(const int* __restrict__ counts, int* __restrict__ starts, int G) {
  if (blockIdx.x == 0 && threadIdx.x == 0) {
    int acc = 0;
    for (int g = 0; g < G; ++g) { starts[g] = acc; acc += counts[g]; }
  }
}
// One block per graph. rank(i) = #{j in graph : last[j] > last[i] or (==, j<i)}
// == exact position under lexsort((-last, batch)) with stable ties.
__global__ void k_sortpool(const float* __restrict__ per_atom, const int* __restrict__ starts,
                           const int* __restrict__ counts, float* __restrict__ pooled) {
  int g = blockIdx.x;
  int st = starts[g], cnt = counts[g];
  for (int li = threadIdx.x; li < cnt; li += blockDim.x) {
    int i = st + li;
    float vi = per_atom[(long)i * 97 + 96];
    int rank = 0;
    for (int lj = 0; lj < cnt; ++lj) {
      int j = st + lj;
      float vj = per_atom[(long)j * 97 + 96];
      rank += (vj > vi) || (vj == vi && j < i);
      if (rank >= 32) break;   // can only grow -> not in top-K
    }
    if (rank < 32) {
      float* dst = pooled + ((long)g * 32 + rank) * 97;
      const float* src = per_atom + (long)i * 97;
      for (int d = 0; d < 97; ++d) dst[d] = src[d];
    }
  }
}

// ----------------------------- conv1 ---------------------------------------
// blockDim = 512: tid -> (k = tid/16, c = tid%16). out1[g][c][k].
__global__ void k_conv1(const float* __restrict__ pooled, const float* __restrict__ w,
                        const float* __restrict__ b, float* __restrict__ out1) {
  int g = blockIdx.x, tid = threadIdx.x;
  int k = tid >> 4, c = tid & 15;
  const float* p = pooled + ((long)g * 32 + k) * 97;
  float acc = b[c];
  for (int d = 0; d < 97; ++d) acc += p[d] * w[c * 97 + d];
  out1[(long)g * 512 + c * 32 + k] = fmaxf(acc, 0.f);
}

// ----------------------------- head ----------------------------------------
// maxpool(2) -> conv2(k=5) -> relu -> lin1 -> relu -> lin2. One block per graph.
__global__ void k_head(const float* __restrict__ out1,
                       const float* __restrict__ c2w, const float* __restrict__ c2b,
                       const float* __restrict__ l1w, const float* __restrict__ l1b,
                       const float* __restrict__ l2w, const float* __restrict__ l2b,
                       float* __restrict__ out) {
  __shared__ float s1[512];    // conv1 out (16 ch x 32)
  __shared__ float pm[256];    // maxpooled (16 ch x 16)
  __shared__ float flat[384];  // conv2 out flattened (32 ch x 12)
  __shared__ float h128[128];
  __shared__ float red[128];
  int g = blockIdx.x, tid = threadIdx.x;
  for (int i = tid; i < 512; i += 256) s1[i] = out1[(long)g * 512 + i];
  __syncthreads();
  {
    int c = tid >> 4, t = tid & 15;
    pm[c * 16 + t] = fmaxf(s1[c * 32 + 2 * t], s1[c * 32 + 2 * t + 1]);
  }
  __syncthreads();
  for (int o = tid; o < 384; o += 256) {
    int c2 = o / 12, pos = o % 12;
    float acc = c2b[c2];
    for (int c1 = 0; c1 < 16; ++c1)
#pragma unroll
      for (int j = 0; j < 5; ++j)
        acc += pm[c1 * 16 + pos + j] * c2w[(c2 * 16 + c1) * 5 + j];
    flat[o] = fmaxf(acc, 0.f);   // flatten order: c2*12 + pos == o
  }
  __syncthreads();
  if (tid < 128) {
    float acc = l1b[tid];
    for (int i = 0; i < 384; ++i) acc += flat[i] * l1w[i * 128 + tid];
    h128[tid] = fmaxf(acc, 0.f);
  }
  __syncthreads();
  if (tid < 128) red[tid] = h128[tid] * l2w[tid];
  __syncthreads();
  for (int s = 64; s > 0; s >>= 1) {
    if (tid < s) red[tid] += red[tid + s];
    __syncthreads();
  }
  if (tid == 0) out[g] = red[0] + l2b[0];
}

// ----------------------------- launch --------------------------------------
extern "C" void kernel_launch(void* const* d_in, const int* in_sizes, int n_in,
                              void* d_out, int out_size, void* d_ws, size_t ws_size,
                              hipStream_t stream) {
  const float* x    = (const float*)d_in[0];
  const int*   ei   = (const int*)d_in[1];
  const int*   batch= (const int*)d_in[2];
  const float* W0   = (const float*)d_in[3];
  const float* b0   = (const float*)d_in[4];
  const float* W1   = (const float*)d_in[5];
  const float* b1   = (const float*)d_in[6];
  const float* W2   = (const float*)d_in[7];
  const float* b2   = (const float*)d_in[8];
  const float* W3   = (const float*)d_in[9];
  const float* b3   = (const float*)d_in[10];
  const float* c1w  = (const float*)d_in[11];
  const float* c1b  = (const float*)d_in[12];
  const float* c2w  = (const float*)d_in[13];
  const float* c2b  = (const float*)d_in[14];
  const float* l1w  = (const float*)d_in[15];
  const float* l1b  = (const float*)d_in[16];
  const float* l2w  = (const float*)d_in[17];
  const float* l2b  = (const float*)d_in[18];

  const int N = in_sizes[0] / 14;
  const int E = in_sizes[1] / 2;
  const int G = out_size - N * 97;   // 256

  float* out      = (float*)d_out;
  float* per_atom = out + G;

  // workspace carve (256B aligned chunks)
  char* wsb = (char*)d_ws;
  size_t off = 0;
  auto carve = [&](size_t bytes) -> void* {
    void* p = wsb + off;
    off += (bytes + 255) & ~(size_t)255;
    return p;
  };
  int*   degi  = (int*)  carve((size_t)N * 4);
  float* dinv  = (float*)carve((size_t)N * 4);
  float* snorm = (float*)carve((size_t)N * 4);
  float* enorm = (float*)carve((size_t)E * 4);
  float* xpad  = (float*)carve((size_t)N * 16 * 4);
  float* hcur  = (float*)carve((size_t)N * 32 * 4);
  float* hw    = (float*)carve((size_t)N * 32 * 4);
  float* agg   = (float*)carve((size_t)N * 32 * 4);
  float* hw4   = (float*)carve((size_t)N * 4);
  float* agg4  = (float*)carve((size_t)N * 4);
  float* Wt    = (float*)carve((size_t)32 * 32 * 4);
  float* pooled= (float*)carve((size_t)G * 32 * 97 * 4);
  float* out1  = (float*)carve((size_t)G * 512 * 4);
  int*   counts= (int*)  carve((size_t)G * 4);
  int*   starts= (int*)  carve((size_t)G * 4);
  (void)ws_size; (void)n_in;

  auto nb = [](long n, int tpb) { return dim3((unsigned)((n + tpb - 1) / tpb)); };
  dim3 B(TPB);

  // degree / normalization
  k_zero_i<<<nb(N, TPB), B, 0, stream>>>(degi, N);
  k_deg  <<<nb(E, TPB), B, 0, stream>>>(ei, degi, E);
  k_dinv <<<nb(N, TPB), B, 0, stream>>>(degi, dinv, snorm, N);
  k_enorm<<<nb(E, TPB), B, 0, stream>>>(ei, dinv, enorm, E);

  // pad x to 16 cols once
  k_pad_x<<<nb((long)N * 16, TPB), B, 0, stream>>>(x, xpad, N);

  // 3 GCN layers of width 32 (WMMA GEMM + edge atomics + tanh combine)
  const float* Warr[3] = {W0, W1, W2};
  const float* barr[3] = {b0, b1, b2};
  long tiles = (long)((N + 15) / 16) * 2;
  for (int l = 0; l < 3; ++l) {
    int KP = (l == 0) ? 16 : 32;
    int Kreal = (l == 0) ? 14 : 32;
    k_zero_f<<<nb((long)N * 32, TPB), B, 0, stream>>>(agg, (long)N * 32);
    k_prep_w<<<nb(32 * KP, TPB), B, 0, stream>>>(Warr[l], Wt, Kreal, KP);
    if (l == 0)
      k_gemm32_wmma<16><<<nb(tiles * 32, TPB), B, 0, stream>>>(xpad, Wt, hw, N);
    else
      k_gemm32_wmma<32><<<nb(tiles * 32, TPB), B, 0, stream>>>(hcur, Wt, hw, N);
    k_edge_agg32<<<nb((long)E * 32, TPB), B, 0, stream>>>(ei, enorm, hw, agg, E);
    k_combine   <<<nb((long)N * 32, TPB), B, 0, stream>>>(agg, hw, snorm, barr[l], hcur,
                                                          per_atom, l * 32, N);
  }

  // layer 3 (32 -> 1)
  k_zero_f   <<<nb(N, TPB), B, 0, stream>>>(agg4, N);
  k_hw4      <<<nb(N, TPB), B, 0, stream>>>(hcur, W3, hw4, N);
  k_edge_agg1<<<nb(E, TPB), B, 0, stream>>>(ei, enorm, hw4, agg4, E);
  k_combine4 <<<nb(N, TPB), B, 0, stream>>>(agg4, hw4, snorm, b3, per_atom, N);

  // sort pooling: zero -> count -> scan -> zero pooled -> rank+scatter
  k_zero_i  <<<nb(G, TPB), B, 0, stream>>>(counts, G);
  k_counts  <<<nb(N, TPB), B, 0, stream>>>(batch, counts, N);
  k_starts  <<<1, 1, 0, stream>>>(counts, starts, G);
  k_zero_f  <<<nb((long)G * 32 * 97, TPB), B, 0, stream>>>(pooled, (long)G * 32 * 97);
  k_sortpool<<<G, B, 0, stream>>>(per_atom, starts, counts, pooled);

  // head
  k_conv1<<<G, 512, 0, stream>>>(pooled, c1w, c1b, out1);
  k_head <<<G, B, 0, stream>>>(out1, c2w, c2b, l1w, l1b, l2w, l2b, out);
}